// SpikeFP64Multiplier_43860206027330
// MI455X (gfx1250) — compile-verified
//
#include <hip/hip_runtime.h>
#include <stdint.h>
#include <stddef.h>

// ---------------------------------------------------------------------------
// SpikeFP64Multiplier for MI455X (gfx1250).
//
// Reference = gate-level IEEE-754 fp64 multiplier on {0,1}-valued f32 bit
// vectors (MSB-first bit layout). All soft gates are exact on {0,1}, so we
// replace the gate network with a bit-exact integer emulation:
//   pack 64 f32 bits -> u64, run the reference's multiply algorithm in
//   integer registers, unpack 64 bits -> f32.
// Memory-bound: ~100.7 MB total traffic -> ~4.3 us floor @ 23.3 TB/s HBM
// (fully L2-resident on graph replay, 192 MB L2). Compute ~500 VALU/elem is
// far below that floor. WMMA does not apply (no matmul structure; f32 accum
// cannot hold 2^52-weighted exact bit sums). CDNA5-specific path used here:
// async tensor-class data movement (global_load_async_to_lds_b128 +
// s_wait_asynccnt) to stage tiles into LDS with dense 128B requests, then
// bank-conflict-free skewed ds_load_b128 reads for per-thread row packing.
// ---------------------------------------------------------------------------

#define AS1 __attribute__((address_space(1)))
#define AS3 __attribute__((address_space(3)))

typedef int v4i __attribute__((vector_size(16)));   // builtin's b128 payload type

#if defined(__HIP_DEVICE_COMPILE__) && \
    __has_builtin(__builtin_amdgcn_global_load_async_to_lds_b128)
#define USE_ASYNC_LDS 1
#else
#define USE_ASYNC_LDS 0
#endif

static constexpr int BLOCK = 64;   // 2 waves (wave32) per block, 1 elem/thread

// ---- bit-exact emulation of the reference fp64_mul ------------------------
__device__ __forceinline__ uint64_t fp64_mul_ref(uint64_t a, uint64_t b) {
  const uint32_t sA = (uint32_t)(a >> 63), sB = (uint32_t)(b >> 63);
  const uint32_t eA = (uint32_t)(a >> 52) & 0x7FFu;
  const uint32_t eB = (uint32_t)(b >> 52) & 0x7FFu;
  const uint64_t mA = a & 0xFFFFFFFFFFFFFULL;
  const uint64_t mB = b & 0xFFFFFFFFFFFFFULL;
  const uint32_t sign = sA ^ sB;

  // significand: implicit bit = (exponent != 0)  (ref: eX_nz)
  const uint64_t MA = mA | ((uint64_t)(eA != 0) << 52);
  const uint64_t MB = mB | ((uint64_t)(eB != 0) << 52);

  // 106-bit product in a 108-bit frame (ref: mul54 / P108)
  const uint64_t plo = MA * MB;
  const uint64_t phi = __umul64hi(MA, MB);

  // leading-zero count within the 108-bit frame (ref: lzd108)
  int lz;
  if (phi)      lz = __builtin_clzll(phi) - 20;   // in [2, 43]
  else if (plo) lz = 44 + __builtin_clzll(plo);   // in [44, 107]
  else          lz = 0;                           // p==0 -> zero_sel anyway

  // normalize: leading 1 -> bit 107 (ref: barrel_left108)
  uint64_t vhi, vlo;
  if (lz >= 64)     { vhi = plo << (lz - 64); vlo = 0; }
  else if (lz == 0) { vhi = phi;              vlo = plo; }
  else              { vhi = (phi << lz) | (plo >> (64 - lz)); vlo = plo << lz; }

  // mant53 = bits 107..55, guard = bit 54, sticky = bits 53..0
  const uint64_t mant53 = (vhi << 9) | (vlo >> 55);
  const uint32_t guard  = (uint32_t)(vlo >> 54) & 1u;
  const uint32_t sticky = (vlo & ((1ULL << 54) - 1)) != 0;

  // round to nearest even
  const uint32_t rnd    = guard & (sticky | ((uint32_t)mant53 & 1u));
  const uint64_t sum    = mant53 + rnd;
  const uint32_t rcarry = (uint32_t)(sum >> 53);
  const uint64_t frac   = sum & ((1ULL << 52) - 1);

  // exponent in 13-bit two's complement; range [-1128, 3075] -> plain int OK
  const int E = (int)eA + (int)eB + 3 - 1023 - lz + (int)rcarry;
  const bool underflow = (E <= 0);       // neg | e_zero
  const bool overflow  = (E >= 2047);    // !neg & (bit11 | low11==0x7FF)

  // specials (priority: nan > inf > zero > normal, sign always sA^sB)
  const bool Anan = (eA == 0x7FF) && (mA != 0);
  const bool Bnan = (eB == 0x7FF) && (mB != 0);
  const bool Ainf = (eA == 0x7FF) && (mA == 0);
  const bool Binf = (eB == 0x7FF) && (mB == 0);
  const bool Azer = (eA == 0) && (mA == 0);
  const bool Bzer = (eB == 0) && (mB == 0);
  const bool nan_sel  = Anan | Bnan | (Ainf & Bzer) | (Binf & Azer);
  const bool inf_sel  = Ainf | Binf | overflow;
  const bool zero_sel = Azer | Bzer | underflow;

  uint32_t eo = (uint32_t)E & 0x7FFu;
  uint64_t fo = frac;
  if (zero_sel) { eo = 0;      fo = 0; }
  if (inf_sel)  { eo = 0x7FFu; fo = 0; }
  if (nan_sel)  { eo = 0x7FFu; fo = 1ULL << 51; }  // canonical qNaN payload

  return ((uint64_t)sign << 63) | ((uint64_t)eo << 52) | fo;
}

// ---- packing helpers ------------------------------------------------------
// input floats are exactly 0.0f (0x00000000) or 1.0f (0x3F800000): bit 29.
__device__ __forceinline__ uint32_t bit_of(uint32_t u) { return (u >> 29) & 1u; }

__device__ __forceinline__ uint64_t pack_row_global(const uint4* __restrict__ g) {
  uint32_t hi = 0, lo = 0;
#pragma unroll
  for (int k = 0; k < 8; ++k) {
    uint4 u = g[k];
    hi |= bit_of(u.x) << (31 - 4 * k);
    hi |= bit_of(u.y) << (30 - 4 * k);
    hi |= bit_of(u.z) << (29 - 4 * k);
    hi |= bit_of(u.w) << (28 - 4 * k);
  }
#pragma unroll
  for (int k = 8; k < 16; ++k) {
    uint4 u = g[k];
    lo |= bit_of(u.x) << (63 - 4 * k);
    lo |= bit_of(u.y) << (62 - 4 * k);
    lo |= bit_of(u.z) << (61 - 4 * k);
    lo |= bit_of(u.w) << (60 - 4 * k);
  }
  return ((uint64_t)hi << 32) | lo;
}

#if USE_ASYNC_LDS
// Skewed chunk order (tid+k)&15 -> the 32 lanes of a wave touch disjoint
// bank groups of the 64-bank LDS instead of all hitting banks 4k..4k+3.
__device__ __forceinline__ uint64_t pack_row_lds(const uint4* s, int tid) {
  uint64_t w = 0;
#pragma unroll
  for (int k = 0; k < 16; ++k) {
    const int c = (tid + k) & 15;
    uint4 u = s[tid * 16 + c];
    const uint32_t nib = (bit_of(u.x) << 3) | (bit_of(u.y) << 2) |
                         (bit_of(u.z) << 1) |  bit_of(u.w);
    w |= (uint64_t)nib << (60 - 4 * c);
  }
  return w;
}
#endif

// ---- kernel ---------------------------------------------------------------
__global__ __launch_bounds__(BLOCK)
void fp64mul_bits_kernel(const uint32_t* __restrict__ A,
                         const uint32_t* __restrict__ B,
                         uint32_t* __restrict__ out, int n) {
  const int tid = threadIdx.x;
  const int e0  = blockIdx.x * BLOCK;
  const int e   = e0 + tid;

  uint64_t aw, bw;

#if USE_ASYNC_LDS
  // Stage the block tile (64 rows x 256B per operand) into LDS with the
  // CDNA5 async data mover: dense coalesced 128b transfers, ASYNCcnt-tracked.
  __shared__ uint4 sA[BLOCK * 16];
  __shared__ uint4 sB[BLOCK * 16];
  {
    const uint4* gA = reinterpret_cast<const uint4*>(A) + (size_t)e0 * 16;
    const uint4* gB = reinterpret_cast<const uint4*>(B) + (size_t)e0 * 16;
#pragma unroll
    for (int k = 0; k < 16; ++k) {
      const int m = tid + k * BLOCK;   // fully coalesced across the block
      __builtin_amdgcn_global_load_async_to_lds_b128(
          (AS1 v4i*)(gA + m), (AS3 v4i*)(sA + m), 0, 0);
      __builtin_amdgcn_global_load_async_to_lds_b128(
          (AS1 v4i*)(gB + m), (AS3 v4i*)(sB + m), 0, 0);
    }
  }
#if __has_builtin(__builtin_amdgcn_s_wait_asynccnt)
  __builtin_amdgcn_s_wait_asynccnt(0);
#else
  asm volatile("s_wait_asynccnt 0x0" ::: "memory");
#endif
  __syncthreads();

  aw = pack_row_lds(sA, tid);
  bw = pack_row_lds(sB, tid);
#else
  if (e >= n) return;
  aw = pack_row_global(reinterpret_cast<const uint4*>(A) + (size_t)e * 16);
  bw = pack_row_global(reinterpret_cast<const uint4*>(B) + (size_t)e * 16);
#endif

  const uint64_t r = fp64_mul_ref(aw, bw);
  const uint32_t rhi = (uint32_t)(r >> 32), rlo = (uint32_t)r;

  // unpack 64 bits -> 64 f32 (0.0 / 1.0), stored as 16 x b128
  uint4* o = reinterpret_cast<uint4*>(out) + (size_t)e * 16;
#pragma unroll
  for (int k = 0; k < 8; ++k) {
    uint4 w;
    w.x = ((rhi >> (31 - 4 * k)) & 1u) * 0x3F800000u;
    w.y = ((rhi >> (30 - 4 * k)) & 1u) * 0x3F800000u;
    w.z = ((rhi >> (29 - 4 * k)) & 1u) * 0x3F800000u;
    w.w = ((rhi >> (28 - 4 * k)) & 1u) * 0x3F800000u;
    o[k] = w;
  }
#pragma unroll
  for (int k = 8; k < 16; ++k) {
    uint4 w;
    w.x = ((rlo >> (63 - 4 * k)) & 1u) * 0x3F800000u;
    w.y = ((rlo >> (62 - 4 * k)) & 1u) * 0x3F800000u;
    w.z = ((rlo >> (61 - 4 * k)) & 1u) * 0x3F800000u;
    w.w = ((rlo >> (60 - 4 * k)) & 1u) * 0x3F800000u;
    o[k] = w;
  }
}

// ---- launcher -------------------------------------------------------------
extern "C" void kernel_launch(void* const* d_in, const int* in_sizes, int n_in,
                              void* d_out, int out_size, void* d_ws, size_t ws_size,
                              hipStream_t stream) {
  const uint32_t* A = (const uint32_t*)d_in[0];   // f32 bits, {0, 0x3F800000}
  const uint32_t* B = (const uint32_t*)d_in[1];
  uint32_t* out = (uint32_t*)d_out;
  const int n = in_sizes[0] / 64;                 // 131072 rows (divisible by 64)

  dim3 grid((n + BLOCK - 1) / BLOCK), block(BLOCK);
  hipLaunchKernelGGL(fp64mul_bits_kernel, grid, block, 0, stream, A, B, out, n);
}